// NN_49993419325985
// MI455X (gfx1250) — compile-verified
//
#include <hip/hip_runtime.h>
#include <hip/hip_bf16.h>

#define SEQ 520
#define EMB 6
#define NH 3
#define TILES 33            // ceil(520/16)
#define NTASK (NH * TILES)  // 99
#define LOG2E 1.4426950408889634f
#define LNEPS 1e-5f

typedef __attribute__((ext_vector_type(2))) float v2f;
typedef __attribute__((ext_vector_type(8))) float v8f;

#ifdef __has_builtin
#if __has_builtin(__builtin_amdgcn_wmma_f32_16x16x4_f32)
#define USE_WMMA_F32X4 1
#endif
#endif
#ifndef USE_WMMA_F32X4
#define USE_WMMA_F32X4 0
#endif

static __device__ __forceinline__ float fast_exp2(float x) {
#if __has_builtin(__builtin_amdgcn_exp2f)
  return __builtin_amdgcn_exp2f(x);
#else
  return __builtin_exp2f(x);
#endif
}

static __device__ __forceinline__ float fast_rsqrt(float x) {
#if __has_builtin(__builtin_amdgcn_rsqf)
  return __builtin_amdgcn_rsqf(x);
#else
  return 1.0f / __builtin_sqrtf(x);
#endif
}

__global__ __launch_bounds__(256) void attn_mlp_fused(
    const float* __restrict__ x, const float* __restrict__ Wq,
    const float* __restrict__ Wk, const float* __restrict__ Wv,
    const float* __restrict__ Wo, const float* __restrict__ ln1_g,
    const float* __restrict__ ln1_b, const float* __restrict__ fc1_w,
    const float* __restrict__ fc1_b, const float* __restrict__ ln2_g,
    const float* __restrict__ ln2_b, const float* __restrict__ fc2_w,
    const float* __restrict__ fc2_b, float* __restrict__ out) {
  const int b = blockIdx.x;
  const int tid = threadIdx.x;

  __shared__ float xs[SEQ];
  __shared__ float sNum[NH * SEQ];
  __shared__ float sDen[NH * SEQ];
  __shared__ float red[EMB][256];
  __shared__ float sQ[EMB], sK[EMB], sV[EMB];
  __shared__ float sC[NH];
  __shared__ float sWvp[NH][EMB];
  __shared__ float sG1[EMB], sB1[EMB];
  __shared__ float sXmax, sXmin;

  // ---- Phase A: load x row, min/max, fold rank-1 weight constants ----
  float lmax = -3.4e38f, lmin = 3.4e38f;
  for (int s = tid; s < SEQ; s += 256) {
    float v = x[b * SEQ + s];
    xs[s] = v;
    lmax = fmaxf(lmax, v);
    lmin = fminf(lmin, v);
  }
  red[0][tid] = lmax;
  red[1][tid] = lmin;
  if (tid < EMB) {
    float q = 0.f, k = 0.f, v = 0.f;
    for (int i = 0; i < EMB; ++i) {
      q += Wq[i * EMB + tid];
      k += Wk[i * EMB + tid];
      v += Wv[i * EMB + tid];
    }
    sQ[tid] = q; sK[tid] = k; sV[tid] = v;
    sG1[tid] = ln1_g[tid]; sB1[tid] = ln1_b[tid];
  }
  __syncthreads();
  if (tid < NH)
    sC[tid] = (sQ[2 * tid] * sK[2 * tid] + sQ[2 * tid + 1] * sK[2 * tid + 1]) *
              0.70710678118654752f;  // includes 1/sqrt(D)
  if (tid < NH * EMB) {
    int h = tid / EMB, e = tid % EMB;
    sWvp[h][e] = sV[2 * h] * Wo[(2 * h) * EMB + e] +
                 sV[2 * h + 1] * Wo[(2 * h + 1) * EMB + e];
  }
  for (int st = 128; st >= 1; st >>= 1) {
    __syncthreads();
    if (tid < st) {
      red[0][tid] = fmaxf(red[0][tid], red[0][tid + st]);
      red[1][tid] = fminf(red[1][tid], red[1][tid + st]);
    }
  }
  __syncthreads();
  if (tid == 0) { sXmax = red[0][0]; sXmin = red[1][0]; }
  __syncthreads();

  // ---- Phase B: softmax numerator/denominator reductions ----
  const int wave = tid >> 5;
  const int lane = tid & 31;
  const int half = lane >> 4;
  const int nn = lane & 15;

#if USE_WMMA_F32X4
  // One wave per (head, 16-row s-tile). exp tile (16x4, f32) x B(4x16)
  // with B columns [1, x_t, 0...] -> D[:,0]=denominator, D[:,1]=numerator.
  for (int task = wave; task < NTASK; task += 8) {
    const int h = task / TILES;
    const int tile = task % TILES;
    const int sb = tile * 16;
    int srow = sb + nn;
    if (srow > SEQ - 1) srow = SEQ - 1;  // clamp (no EXEC divergence)
    const float a = xs[srow] * sC[h];
    const float m = (a > 0.f) ? a * sXmax : a * sXmin;  // exact row max
    const float al2 = a * LOG2E;
    const float ml2 = m * LOG2E;
    const int K0 = 2 * half;  // A/B K for VGPR j is j + 2*half
    v8f acc = {};
    for (int k0 = 0; k0 < SEQ; k0 += 4) {
      float xt0 = xs[k0 + K0];
      float xt1 = xs[k0 + K0 + 1];
      v2f A, B;
      A.x = fast_exp2(al2 * xt0 - ml2);
      A.y = fast_exp2(al2 * xt1 - ml2);
      B.x = (nn == 0) ? 1.0f : ((nn == 1) ? xt0 : 0.0f);
      B.y = (nn == 0) ? 1.0f : ((nn == 1) ? xt1 : 0.0f);
      acc = __builtin_amdgcn_wmma_f32_16x16x4_f32(false, A, false, B,
                                                  (short)0, acc, false, false);
    }
    if (nn <= 1) {
#pragma unroll
      for (int r = 0; r < 8; ++r) {
        int srec = sb + 8 * half + r;  // D VGPR r: rows r (lanes<16), r+8
        if (srec < SEQ) {
          if (nn == 0) sDen[h * SEQ + srec] = acc[r];
          else         sNum[h * SEQ + srec] = acc[r];
        }
      }
    }
  }
#else
  for (int idx = tid; idx < NH * SEQ; idx += 256) {
    int h = idx / SEQ, s = idx % SEQ;
    float a = xs[s] * sC[h];
    float m = (a > 0.f) ? a * sXmax : a * sXmin;
    float al2 = a * LOG2E, ml2 = m * LOG2E;
    float den = 0.f, num = 0.f;
    for (int t = 0; t < SEQ; ++t) {
      float ev = fast_exp2(al2 * xs[t] - ml2);
      den += ev;
      num += ev * xs[t];
    }
    sDen[idx] = den;
    sNum[idx] = num;
  }
#endif
  __syncthreads();

  // ---- Phase C: LN1 + fc1 partial dot, deterministic tree reduction ----
  float part[EMB] = {0.f, 0.f, 0.f, 0.f, 0.f, 0.f};
  for (int s = tid; s < SEQ; s += 256) {
    float f0 = sNum[0 * SEQ + s] / sDen[0 * SEQ + s];
    float f1 = sNum[1 * SEQ + s] / sDen[1 * SEQ + s];
    float f2 = sNum[2 * SEQ + s] / sDen[2 * SEQ + s];
    float o[EMB];
    float mean = 0.f;
#pragma unroll
    for (int e = 0; e < EMB; ++e) {
      o[e] = f0 * sWvp[0][e] + f1 * sWvp[1][e] + f2 * sWvp[2][e];
      mean += o[e];
    }
    mean *= (1.0f / EMB);
    float var = 0.f;
#pragma unroll
    for (int e = 0; e < EMB; ++e) { float d = o[e] - mean; var += d * d; }
    var *= (1.0f / EMB);
    float rs = fast_rsqrt(var + LNEPS);
    float xn[EMB];
#pragma unroll
    for (int e = 0; e < EMB; ++e) xn[e] = (o[e] - mean) * rs * sG1[e] + sB1[e];
    const float* fw = fc1_w + s * EMB;
#pragma unroll
    for (int ep = 0; ep < EMB; ++ep) {
      const float* row = fw + ep * (SEQ * EMB);
      float acc2 = 0.f;
#pragma unroll
      for (int e = 0; e < EMB; ++e) acc2 += xn[e] * row[e];
      part[ep] += acc2;
    }
  }
#pragma unroll
  for (int e = 0; e < EMB; ++e) red[e][tid] = part[e];
  for (int st = 128; st >= 1; st >>= 1) {
    __syncthreads();
    if (tid < st) {
#pragma unroll
      for (int e = 0; e < EMB; ++e) red[e][tid] += red[e][tid + st];
    }
  }
  __syncthreads();

  // ---- Phase D: bias+relu, LN2, fc2, softmax(4) ----
  if (tid == 0) {
    float y[EMB];
    float mean = 0.f;
#pragma unroll
    for (int e = 0; e < EMB; ++e) {
      float v = red[e][0] + fc1_b[e];
      y[e] = v > 0.f ? v : 0.f;
      mean += y[e];
    }
    mean *= (1.0f / EMB);
    float var = 0.f;
#pragma unroll
    for (int e = 0; e < EMB; ++e) { float d = y[e] - mean; var += d * d; }
    var *= (1.0f / EMB);
    float rs = fast_rsqrt(var + LNEPS);
    float xn[EMB];
#pragma unroll
    for (int e = 0; e < EMB; ++e)
      xn[e] = (y[e] - mean) * rs * ln2_g[e] + ln2_b[e];
    float o2[4];
    float mx = -3.4e38f;
#pragma unroll
    for (int k = 0; k < 4; ++k) {
      float acc2 = fc2_b[k];
#pragma unroll
      for (int e = 0; e < EMB; ++e) acc2 += xn[e] * fc2_w[k * EMB + e];
      o2[k] = acc2;
      mx = fmaxf(mx, acc2);
    }
    float se = 0.f;
#pragma unroll
    for (int k = 0; k < 4; ++k) {
      o2[k] = fast_exp2((o2[k] - mx) * LOG2E);
      se += o2[k];
    }
    float inv = 1.0f / se;
#pragma unroll
    for (int k = 0; k < 4; ++k) out[b * 4 + k] = o2[k] * inv;
  }
}

extern "C" void kernel_launch(void* const* d_in, const int* in_sizes, int n_in,
                              void* d_out, int out_size, void* d_ws, size_t ws_size,
                              hipStream_t stream) {
  (void)n_in; (void)d_ws; (void)ws_size; (void)out_size;
  const float* x     = (const float*)d_in[0];
  const float* Wq    = (const float*)d_in[1];
  const float* Wk    = (const float*)d_in[2];
  const float* Wv    = (const float*)d_in[3];
  const float* Wo    = (const float*)d_in[4];
  const float* ln1_g = (const float*)d_in[5];
  const float* ln1_b = (const float*)d_in[6];
  const float* fc1_w = (const float*)d_in[7];
  const float* fc1_b = (const float*)d_in[8];
  const float* ln2_g = (const float*)d_in[9];
  const float* ln2_b = (const float*)d_in[10];
  const float* fc2_w = (const float*)d_in[11];
  const float* fc2_b = (const float*)d_in[12];
  float* out = (float*)d_out;
  const int B = in_sizes[0] / SEQ;
  attn_mlp_fused<<<B, 256, 0, stream>>>(x, Wq, Wk, Wv, Wo, ln1_g, ln1_b, fc1_w,
                                        fc1_b, ln2_g, ln2_b, fc2_w, fc2_b, out);
}